// SCNN_71485435675281
// MI455X (gfx1250) — compile-verified
//
#include <hip/hip_runtime.h>

typedef __attribute__((ext_vector_type(2))) float v2f;
typedef __attribute__((ext_vector_type(8))) float v8f;

#define MP 64  // L rows per m-chunk in pass 1

// ---------------------------------------------------------------------------
// Pass 1 (bandwidth-bound, streams all of L exactly once):
//   v_i[k, n] = sum_m Wf_i[k, m] * L_i[m, n]        (K=4 x N, fp32)
// Each thread owns 4 consecutive columns (b128 loads), each block a 64-row
// chunk; partial sums merged with global_atomic_add_f32. grid.z = branch.
// ---------------------------------------------------------------------------
__global__ void __launch_bounds__(128)
wfL_kernel(const float* __restrict__ L0, const float* __restrict__ Wf0, float* __restrict__ v0,
           const float* __restrict__ L1, const float* __restrict__ Wf1, float* __restrict__ v1,
           const float* __restrict__ L2, const float* __restrict__ Wf2, float* __restrict__ v2)
{
    const float* L; const float* Wf; float* v; int N;
    if (blockIdx.z == 0)      { L = L0; Wf = Wf0; v = v0; N = 4096; }
    else if (blockIdx.z == 1) { L = L1; Wf = Wf1; v = v1; N = 8192; }
    else                      { L = L2; Wf = Wf2; v = v2; N = 4096; }

    int n0 = (blockIdx.x * 128 + threadIdx.x) * 4;
    int m0 = blockIdx.y * MP;
    if (n0 >= N || m0 >= N) return;

    float acc[4][4] = {};
#pragma unroll 4
    for (int m = m0; m < m0 + MP; ++m) {
        float4 Lv = *(const float4*)(L + (size_t)m * N + n0);
        // read-once stream: branch-free prefetch of a future row (clamped in-bounds)
        int mp = m + 8; mp = (mp < N - 1) ? mp : (N - 1);
        __builtin_prefetch(L + (size_t)mp * N + n0, 0, 1);
        float e[4] = {Lv.x, Lv.y, Lv.z, Lv.w};
        float w[4];
#pragma unroll
        for (int k = 0; k < 4; ++k) w[k] = Wf[k * N + m];   // wave-uniform -> s_load
#pragma unroll
        for (int k = 0; k < 4; ++k)
#pragma unroll
            for (int j = 0; j < 4; ++j)
                acc[k][j] = fmaf(w[k], e[j], acc[k][j]);
    }
#pragma unroll
    for (int k = 0; k < 4; ++k)
#pragma unroll
        for (int j = 0; j < 4; ++j)
            atomicAdd(&v[k * N + n0 + j], acc[k][j]);
}

// ---------------------------------------------------------------------------
// Pass 2 (tiny): z[b,k,f] = sum_n v[k,n] * x[b,n,f]; then
//   hcat[b, off+f] = sum_k relu(z[b,k,f]).
// One block per branch; 64 register accumulators per thread, wave32 shuffle
// tree + LDS cross-wave reduction.
// ---------------------------------------------------------------------------
__global__ void __launch_bounds__(256)
zred_kernel(const float* __restrict__ x0, const float* __restrict__ v0,
            const float* __restrict__ x1, const float* __restrict__ v1,
            const float* __restrict__ x2, const float* __restrict__ v2,
            float* __restrict__ hcat)
{
    const float* x; const float* v; int N, off;
    if (blockIdx.x == 0)      { x = x0; v = v0; N = 4096; off = 0;  }
    else if (blockIdx.x == 1) { x = x1; v = v1; N = 8192; off = 8;  }
    else                      { x = x2; v = v2; N = 4096; off = 16; }

    float acc[2][4][8] = {};
    for (int n = threadIdx.x; n < N; n += 256) {
        float vk[4];
#pragma unroll
        for (int k = 0; k < 4; ++k) vk[k] = v[k * N + n];
#pragma unroll
        for (int b = 0; b < 2; ++b) {
            const float4* xp = (const float4*)(x + ((size_t)b * N + n) * 8);
            float4 xa = xp[0], xb = xp[1];
            float xf[8] = {xa.x, xa.y, xa.z, xa.w, xb.x, xb.y, xb.z, xb.w};
#pragma unroll
            for (int k = 0; k < 4; ++k)
#pragma unroll
                for (int f = 0; f < 8; ++f)
                    acc[b][k][f] = fmaf(vk[k], xf[f], acc[b][k][f]);
        }
    }
    // wave32 tree reduction of all 64 accumulators
#pragma unroll
    for (int b = 0; b < 2; ++b)
#pragma unroll
        for (int k = 0; k < 4; ++k)
#pragma unroll
            for (int f = 0; f < 8; ++f) {
                float t = acc[b][k][f];
                for (int o = 16; o > 0; o >>= 1) t += __shfl_down(t, o, 32);
                acc[b][k][f] = t;
            }

    __shared__ float red[8][64];
    __shared__ float ztot[64];
    int lane = threadIdx.x & 31, wv = threadIdx.x >> 5;
    if (lane == 0) {
#pragma unroll
        for (int b = 0; b < 2; ++b)
#pragma unroll
            for (int k = 0; k < 4; ++k)
#pragma unroll
                for (int f = 0; f < 8; ++f)
                    red[wv][b * 32 + k * 8 + f] = acc[b][k][f];
    }
    __syncthreads();
    if (threadIdx.x < 64) {
        float s = 0.f;
#pragma unroll
        for (int w = 0; w < 8; ++w) s += red[w][threadIdx.x];
        ztot[threadIdx.x] = s;
    }
    __syncthreads();
    if (threadIdx.x < 16) {
        int b = threadIdx.x >> 3, f = threadIdx.x & 7;
        float h = 0.f;
#pragma unroll
        for (int k = 0; k < 4; ++k) h += fmaxf(ztot[b * 32 + k * 8 + f], 0.f);
        hcat[b * 24 + off + f] = h;
    }
}

// ---------------------------------------------------------------------------
// Pass 3: fp32 MLP on V_WMMA_F32_16X16X4_F32 (16x4 A, 4x16 B, 16x16 C).
// M = batch rows (2 valid). Activation buffers are padded to the full 16
// WMMA rows in LDS, with rows 2..15 zeroed once at start, so the A-fragment
// is one unconditional ds_load_b64 (no EXEC-masked scalar DS loads).
// A layout (ISA 7.12.2, 32-bit A 16x4): lanes 0-15 -> K=k,k+1;
// lanes 16-31 -> K=k+2,k+3; m = lane&15. B mirrors the K split per VGPR.
// D: row M=j in c[j] for lanes 0-15 -> batch rows are c[0], c[1].
// ---------------------------------------------------------------------------
__global__ void __launch_bounds__(1024)
mlp_kernel(const float* __restrict__ hcat,
           const float* __restrict__ Win,  const float* __restrict__ b_in,
           const float* __restrict__ Wh,   const float* __restrict__ b_h,
           const float* __restrict__ Wout, const float* __restrict__ b_out,
           float* __restrict__ out)
{
    __shared__ float h0p[16 * 24];     // padded [16,24], rows 2..15 = 0
    __shared__ float hA[16 * 512];     // padded [16,512], rows 2..15 = 0
    __shared__ float hB[16 * 512];     // padded [16,512], rows 2..15 = 0
    int tid  = threadIdx.x;
    int lane = tid & 31;
    int wv   = tid >> 5;       // 0..31
    int half = lane >> 4;      // 0: lanes 0-15, 1: lanes 16-31
    int l15  = lane & 15;

    // zero padded rows once
    for (int i = tid; i < 16 * 24; i += 1024) h0p[i] = 0.f;
    for (int i = tid; i < 16 * 512; i += 1024) { hA[i] = 0.f; hB[i] = 0.f; }
    __syncthreads();
    if (tid < 48) h0p[tid] = hcat[tid];   // rows 0,1 of [16,24]
    __syncthreads();

    // layer 1: [2,24] @ [24,512] + b, relu
    {
        int n0 = wv * 16;
        v8f c = {};
#pragma unroll
        for (int k = 0; k < 24; k += 4) {
            int ka = k + half * 2;                       // even -> 8B aligned
            v2f a = *(const v2f*)&h0p[l15 * 24 + ka];    // ds_load_b64
            v2f b;
            b.x = Win[ka * 512 + n0 + l15];
            b.y = Win[(ka + 1) * 512 + n0 + l15];
            c = __builtin_amdgcn_wmma_f32_16x16x4_f32(false, a, false, b,
                                                      (short)0, c, false, false);
        }
        if (half == 0) {
            float bb = b_in[n0 + l15];
            hA[0 * 512 + n0 + l15] = fmaxf(c[0] + bb, 0.f);
            hA[1 * 512 + n0 + l15] = fmaxf(c[1] + bb, 0.f);
        }
    }
    __syncthreads();

    // layer 2: [2,512] @ [512,512] + b, relu
    {
        int n0 = wv * 16;
        v8f c = {};
        for (int k = 0; k < 512; k += 4) {
            int ka = k + half * 2;
            v2f a = *(const v2f*)&hA[l15 * 512 + ka];    // ds_load_b64
            v2f b;
            b.x = Wh[ka * 512 + n0 + l15];
            b.y = Wh[(ka + 1) * 512 + n0 + l15];
            c = __builtin_amdgcn_wmma_f32_16x16x4_f32(false, a, false, b,
                                                      (short)0, c, false, false);
        }
        if (half == 0) {
            float bb = b_h[n0 + l15];
            hB[0 * 512 + n0 + l15] = fmaxf(c[0] + bb, 0.f);
            hB[1 * 512 + n0 + l15] = fmaxf(c[1] + bb, 0.f);
        }
    }
    __syncthreads();

    // layer 3: [2,512] @ [512,64] + b   (waves 0-3 only; wave-uniform branch)
    if (wv < 4) {
        int n0 = wv * 16;
        v8f c = {};
        for (int k = 0; k < 512; k += 4) {
            int ka = k + half * 2;
            v2f a = *(const v2f*)&hB[l15 * 512 + ka];    // ds_load_b64
            v2f b;
            b.x = Wout[ka * 64 + n0 + l15];
            b.y = Wout[(ka + 1) * 64 + n0 + l15];
            c = __builtin_amdgcn_wmma_f32_16x16x4_f32(false, a, false, b,
                                                      (short)0, c, false, false);
        }
        if (half == 0) {
            float bb = b_out[n0 + l15];
            out[0 * 64 + n0 + l15] = c[0] + bb;
            out[1 * 64 + n0 + l15] = c[1] + bb;
        }
    }
}

// ---------------------------------------------------------------------------
extern "C" void kernel_launch(void* const* d_in, const int* in_sizes, int n_in,
                              void* d_out, int out_size, void* d_ws, size_t ws_size,
                              hipStream_t stream)
{
    // dict order: x0,L0,Wf0, x1,L1,Wf1, x2,L2,Wf2, Win,b_in, Wh,b_h, Wout,b_out
    const float* x0   = (const float*)d_in[0];
    const float* L0   = (const float*)d_in[1];
    const float* Wf0  = (const float*)d_in[2];
    const float* x1   = (const float*)d_in[3];
    const float* L1   = (const float*)d_in[4];
    const float* Wf1  = (const float*)d_in[5];
    const float* x2   = (const float*)d_in[6];
    const float* L2   = (const float*)d_in[7];
    const float* Wf2  = (const float*)d_in[8];
    const float* Win  = (const float*)d_in[9];
    const float* b_in = (const float*)d_in[10];
    const float* Wh   = (const float*)d_in[11];
    const float* b_h  = (const float*)d_in[12];
    const float* Wout = (const float*)d_in[13];
    const float* b_out= (const float*)d_in[14];
    float* out = (float*)d_out;

    float* ws  = (float*)d_ws;
    float* v0  = ws;                 // 4*4096
    float* v1  = v0 + 4 * 4096;      // 4*8192
    float* v2  = v1 + 4 * 8192;      // 4*4096
    float* hc  = v2 + 4 * 4096;      // 48 floats ([2,24] concat)

    // zero the V accumulators (atomicAdd targets) every call
    hipMemsetAsync(ws, 0, (size_t)(4 * (4096 + 8192 + 4096)) * sizeof(float), stream);

    // pass 1: grid sized for the largest branch; small branches early-exit
    dim3 g1(8192 / 512, 8192 / MP, 3);
    wfL_kernel<<<g1, 128, 0, stream>>>(L0, Wf0, v0, L1, Wf1, v1, L2, Wf2, v2);

    // pass 2: one block per branch
    zred_kernel<<<3, 256, 0, stream>>>(x0, v0, x1, v1, x2, v2, hc);

    // pass 3: WMMA MLP, one workgroup of 32 wave32 waves
    mlp_kernel<<<1, 1024, 0, stream>>>(hc, Win, b_in, Wh, b_h, Wout, b_out, out);
}